// RoIAlign_72962904424516
// MI455X (gfx1250) — compile-verified
//
#include <hip/hip_runtime.h>

typedef float v2f __attribute__((ext_vector_type(2)));
typedef float v8f __attribute__((ext_vector_type(8)));

#define OUT_HW 7
#define NCOR   28          // 14 samples/axis * 2 corners
#define CDIM   256
#define HDIM   128
#define WDIM   128
#define SCALE  0.0625f
#define FSTR   33          // padded LDS stride for F (32x32 window)
#define TSTR   17          // padded LDS stride for T (32x16 intermediate)
#define ASTR   33          // padded LDS stride for Ay'
#define WAVES  8

// Separable RoIAlign:  out(7x7) = Ay'(7x28) * F(28x28) * Bx'(28x7), all padded
// to 16/32 and computed with exact-f32 V_WMMA_F32_16X16X4_F32 accumulation.
__global__ __launch_bounds__(256) void roialign_wmma_kernel(
    const float* __restrict__ feat, const float* __restrict__ rois,
    float* __restrict__ out, int K)
{
    __shared__ int   sCy[32], sCx[32];
    __shared__ float sAy[16 * ASTR];            // Ay' (7x28 in 16x32), row-major
    __shared__ float sBx[32 * 16];              // Bx' (28x7 in 32x16), row-major
    __shared__ float sF[WAVES][32 * FSTR];      // per-wave gathered window
    __shared__ float sT[WAVES][32 * TSTR];      // per-wave GEMM1 result

    const int k    = blockIdx.x;
    const int tid  = threadIdx.x;
    const int wave = tid >> 5;
    const int lane = tid & 31;
    const int c    = blockIdx.y * WAVES + wave;

    // ---- zero-init weight matrices (padding must be 0) ----
    #pragma unroll
    for (int i = tid; i < 16 * ASTR; i += 256) sAy[i] = 0.f;
    #pragma unroll
    for (int i = tid; i < 32 * 16;   i += 256) sBx[i] = 0.f;
    __syncthreads();

    // ---- per-ROI corner indices + separable weights (threads 0..31) ----
    if (tid < 32) {
        const float* r = rois + (size_t)k * 5;
        float x1 = r[1] * SCALE - 0.5f, y1 = r[2] * SCALE - 0.5f;
        float x2 = r[3] * SCALE - 0.5f, y2 = r[4] * SCALE - 0.5f;
        float bw = (x2 - x1) * (1.f / OUT_HW);
        float bh = (y2 - y1) * (1.f / OUT_HW);
        int L = tid;
        if (L < NCOR) {
            int q  = L >> 1;           // sample index 0..13
            int hi = L & 1;            // 0: low corner, 1: high corner
            // y axis: Y = y1 + (q+0.5)/2 * bin_h  (exact same f32 value as ref)
            float Y  = y1 + (q + 0.5f) * 0.5f * bh;
            float vy = (Y > -1.f && Y < (float)HDIM) ? 1.f : 0.f;
            float Yc = fmaxf(Y, 0.f);
            int   y0 = min((int)floorf(Yc), HDIM - 1);
            float ly = fminf(fmaxf(Yc - (float)y0, 0.f), 1.f);
            sCy[L] = hi ? min(y0 + 1, HDIM - 1) : y0;
            sAy[(L >> 2) * ASTR + L] = 0.5f * (hi ? ly : 1.f - ly) * vy;
            // x axis
            float X  = x1 + (q + 0.5f) * 0.5f * bw;
            float vx = (X > -1.f && X < (float)WDIM) ? 1.f : 0.f;
            float Xc = fmaxf(X, 0.f);
            int   x0 = min((int)floorf(Xc), WDIM - 1);
            float lx = fminf(fmaxf(Xc - (float)x0, 0.f), 1.f);
            sCx[L] = hi ? min(x0 + 1, WDIM - 1) : x0;
            sBx[L * 16 + (L >> 2)] = 0.5f * (hi ? lx : 1.f - lx) * vx;
        } else {
            sCy[L] = 0; sCx[L] = 0;    // padded rows/cols (weights stay 0)
        }
    }
    __syncthreads();

    // ---- gather 28x28 (padded 32x32) window for this wave's channel ----
    // lane = column -> loads within one feature row are spatially local.
    const int b = (int)rois[(size_t)k * 5];
    const float* fbase = feat + ((size_t)(b * CDIM + c)) * (HDIM * WDIM);
    const int xc = sCx[lane];
    #pragma unroll 4
    for (int rr = 0; rr < 32; ++rr) {
        sF[wave][rr * FSTR + lane] = fbase[sCy[rr] * WDIM + xc];
    }

    const int halfSel = (lane >> 4) << 1;   // K offset: 0 (lanes 0-15) / 2
    const int mrow    = lane & 15;

    // ---- GEMM1: T(32x16) = F(32x32) x Bx'(32x16), exact f32 WMMA ----
    v8f accT[2];
    #pragma unroll
    for (int mt = 0; mt < 2; ++mt) {
        v8f acc = {};
        #pragma unroll
        for (int kk = 0; kk < 8; ++kk) {
            int kb = kk * 4 + halfSel;
            v2f a, bf;
            a.x  = sF[wave][(mt * 16 + mrow) * FSTR + kb];
            a.y  = sF[wave][(mt * 16 + mrow) * FSTR + kb + 1];
            bf.x = sBx[kb * 16 + mrow];
            bf.y = sBx[(kb + 1) * 16 + mrow];
            acc = __builtin_amdgcn_wmma_f32_16x16x4_f32(
                false, a, false, bf, (short)0, acc, false, false);
        }
        accT[mt] = acc;
    }

    // spill T (in C/D layout: VGPR i -> row i (lanes 0-15) / i+8 (lanes 16-31))
    {
        int drow0 = (lane >> 4) * 8;
        #pragma unroll
        for (int mt = 0; mt < 2; ++mt)
            #pragma unroll
            for (int i = 0; i < 8; ++i)
                sT[wave][(mt * 16 + drow0 + i) * TSTR + mrow] = accT[mt][i];
    }

    // ---- GEMM2: out(16x16) = Ay'(16x32) x T(32x16) ----
    v8f acc2 = {};
    #pragma unroll
    for (int kk = 0; kk < 8; ++kk) {
        int kb = kk * 4 + halfSel;
        v2f a, bf;
        a.x  = sAy[mrow * ASTR + kb];
        a.y  = sAy[mrow * ASTR + kb + 1];
        bf.x = sT[wave][kb * TSTR + mrow];
        bf.y = sT[wave][(kb + 1) * TSTR + mrow];
        acc2 = __builtin_amdgcn_wmma_f32_16x16x4_f32(
            false, a, false, bf, (short)0, acc2, false, false);
    }

    // ---- store 7x7 block of D ----
    float* obase = out + ((size_t)(k * CDIM + c)) * (OUT_HW * OUT_HW);
    int orow0 = (lane >> 4) * 8;
    #pragma unroll
    for (int i = 0; i < 8; ++i) {
        int ph = orow0 + i;
        if (ph < OUT_HW && mrow < OUT_HW)
            obase[ph * OUT_HW + mrow] = acc2[i];
    }
}

extern "C" void kernel_launch(void* const* d_in, const int* in_sizes, int n_in,
                              void* d_out, int out_size, void* d_ws, size_t ws_size,
                              hipStream_t stream) {
    const float* feat = (const float*)d_in[0];
    const float* rois = (const float*)d_in[1];
    float* out = (float*)d_out;
    const int K = in_sizes[1] / 5;               // 1024 ROIs
    dim3 grid(K, CDIM / WAVES, 1);               // (1024, 32)
    roialign_wmma_kernel<<<grid, dim3(256, 1, 1), 0, stream>>>(feat, rois, out, K);
}